// SSKernelDiag_27616639713728
// MI455X (gfx1250) — compile-verified
//
#include <hip/hip_runtime.h>
#include <math.h>

// S4D log-Vandermonde kernel, fully fused for MI455X (gfx1250):
//   K[h, 16j+i] = Re( sum_n P[j,n] * Q[n,i] ),
//   P[j,n] = 2*Cb[h,n]*exp(dtA[h,n]*16j),  Q[n,i] = exp(dtA[h,n]*i)
// Per head: (256x32)x(32x16) complex matmul -> two real f32 WMMA chains
// (V_WMMA_F32_16X16X4_F32, K=32 accumulated over 8 steps of 4).
// One workgroup (4 wave32) per head. Transcendentals only in setup:
// across its tiles each wave advances P by the constant complex factor
// S[n] = exp(dtA[n]*1024) (t += nwaves, 16*16*nwaves = 1024), so the
// steady-state loop is pure WMMA + complex-FMA recurrence + stores.

typedef __attribute__((ext_vector_type(2))) float v2f;
typedef __attribute__((ext_vector_type(8))) float v8f;

#define SSD_N2 32

__global__ __launch_bounds__(128)
void s4d_vandermonde_wmma(const float* __restrict__ C_ri,        // (1,H,N2,2)
                          const float* __restrict__ log_dt,      // (H)
                          const float* __restrict__ B_ri,        // (n_ssm,N2,2)
                          const float* __restrict__ inv_A_real,  // (n_ssm,N2)
                          const float* __restrict__ A_imag,      // (n_ssm,N2)
                          float* __restrict__ Kout,              // (H,L)
                          int n_ssm, int L) {
  __shared__ float s_dre[SSD_N2];  // Re(dtA[n])
  __shared__ float s_dim[SSD_N2];  // Im(dtA[n])
  __shared__ float s_cbr[SSD_N2];  // Re(2*Cb[n])
  __shared__ float s_cbi[SSD_N2];  // Im(2*Cb[n])

  const int h      = blockIdx.x;
  const int tid    = threadIdx.x;
  const int lane   = tid & 31;
  const int wave   = tid >> 5;
  const int nwaves = blockDim.x >> 5;

  // ---- Per-head discretization scalars (32 threads, one n each) ----
  if (tid < SSD_N2) {
    const int n  = tid;
    const int hs = h % n_ssm;                 // tile('t n -> (v t) n')
    const float dt  = expf(log_dt[h]);
    const float Are = -expf(inv_A_real[hs * SSD_N2 + n]);
    const float Aim = A_imag[hs * SSD_N2 + n];
    const float dre = Are * dt;
    const float dim = Aim * dt;
    // w = exp(dtA)
    float sw, cw;
    sincosf(dim, &sw, &cw);
    const float e  = expf(dre);
    const float wr = e * cw;
    const float wi = e * sw;
    // Cb = (B*C) * (exp(dtA)-1)/A, folded with the final factor 2
    const float Br = B_ri[(hs * SSD_N2 + n) * 2 + 0];
    const float Bi = B_ri[(hs * SSD_N2 + n) * 2 + 1];
    const float Cr = C_ri[(h * SSD_N2 + n) * 2 + 0];
    const float Ci = C_ri[(h * SSD_N2 + n) * 2 + 1];
    const float BCr = Br * Cr - Bi * Ci;
    const float BCi = Br * Ci + Bi * Cr;
    const float nr = wr - 1.0f, ni = wi;
    const float inv_den = 1.0f / (Are * Are + Aim * Aim);
    const float qr = (nr * Are + ni * Aim) * inv_den;   // (w-1)*conj(A)/|A|^2
    const float qi = (ni * Are - nr * Aim) * inv_den;
    s_dre[n] = dre;
    s_dim[n] = dim;
    s_cbr[n] = 2.0f * (BCr * qr - BCi * qi);
    s_cbi[n] = 2.0f * (BCr * qi + BCi * qr);
  }
  __syncthreads();

  const int   i  = lane & 15;   // output column i == A-matrix row M
  const int   hf = lane >> 4;   // lane half selects K pair in A/B operands
  const int   c0 = 2 * hf;
  const float fi = (float)i;

  // Wave-local constants for the P recurrence:
  const float sc0  = (float)(256 * wave + 16 * i);  // initial 16*j
  const float step = (float)(256 * nwaves);         // 16*j advance per tile

  // Register state (all chunk arrays fully unrolled into VGPRs):
  v2f Qr[8], Qi[8];   // B operand  Q[n,i] = exp(dtA[n]*i)       (constant)
  v2f Pr[8], Pn[8];   // A operand: Pr = Re(P), Pn = -Im(P)      (recurrent)
  v2f Sr[8], Si[8];   // per-tile multiplier S[n] = exp(dtA[n]*step)

#pragma unroll
  for (int k = 0; k < 8; ++k) {
    const int n0 = 4 * k + c0;   // lanes 0-15: K={0,1}; lanes 16-31: K={2,3}
    const int n1 = n0 + 1;
    const float d0r = s_dre[n0], d0i = s_dim[n0];
    const float d1r = s_dre[n1], d1i = s_dim[n1];
    const float b0r = s_cbr[n0], b0i = s_cbi[n0];
    const float b1r = s_cbr[n1], b1i = s_cbi[n1];

    float s0, c0v, s1, c1v;
    // Q[n,i] = exp(dtA[n]*i); B layout: VGPR r + 2*half -> contraction row
    sincosf(d0i * fi, &s0, &c0v);
    sincosf(d1i * fi, &s1, &c1v);
    const float eq0 = expf(d0r * fi);
    const float eq1 = expf(d1r * fi);
    Qr[k] = (v2f){eq0 * c0v, eq1 * c1v};
    Qi[k] = (v2f){eq0 * s0, eq1 * s1};

    // P init at 16*j = sc0:  P = 2*Cb * exp(dtA*sc0)
    sincosf(d0i * sc0, &s0, &c0v);
    sincosf(d1i * sc0, &s1, &c1v);
    const float ep0 = expf(d0r * sc0);
    const float ep1 = expf(d1r * sc0);
    Pr[k] = (v2f){ep0 * (b0r * c0v - b0i * s0), ep1 * (b1r * c1v - b1i * s1)};
    Pn[k] = (v2f){-ep0 * (b0r * s0 + b0i * c0v), -ep1 * (b1r * s1 + b1i * c1v)};

    // Tile-step multiplier S = exp(dtA*step)
    sincosf(d0i * step, &s0, &c0v);
    sincosf(d1i * step, &s1, &c1v);
    const float es0 = expf(d0r * step);
    const float es1 = expf(d1r * step);
    Sr[k] = (v2f){es0 * c0v, es1 * c1v};
    Si[k] = (v2f){es0 * s0, es1 * s1};
  }

  const int ntiles = L >> 8;  // 16x16 output tiles per head (L/256)

  for (int t = wave; t < ntiles; t += nwaves) {
    // Two independent WMMA accumulator chains (re and im contributions)
    v8f acc1 = {};
    v8f acc2 = {};
#pragma unroll
    for (int k = 0; k < 8; ++k) {
      acc1 = __builtin_amdgcn_wmma_f32_16x16x4_f32(
          false, Pr[k], false, Qr[k], (short)0, acc1, false, false);
      acc2 = __builtin_amdgcn_wmma_f32_16x16x4_f32(
          false, Pn[k], false, Qi[k], (short)0, acc2, false, false);
    }
    // D layout: VGPR r, lane-half hf -> M = r + 8*hf, N = i.  l = 256t+16M+i.
    float* outp = Kout + (size_t)h * (size_t)L + (size_t)(256 * t) + i;
#pragma unroll
    for (int r = 0; r < 8; ++r) {
      const int M = r + 8 * hf;
      outp[16 * M] = acc1[r] + acc2[r];
    }
    // Advance P by S (complex mul with Pn = -Im(P) folded in):
    //   Pr' = Pr*Sr + Pn*Si ;  Pn' = Pn*Sr - Pr*Si
#pragma unroll
    for (int k = 0; k < 8; ++k) {
      const v2f pr = Pr[k], pn = Pn[k];
      Pr[k] = pr * Sr[k] + pn * Si[k];
      Pn[k] = pn * Sr[k] - pr * Si[k];
    }
  }
}

extern "C" void kernel_launch(void* const* d_in, const int* in_sizes, int n_in,
                              void* d_out, int out_size, void* d_ws, size_t ws_size,
                              hipStream_t stream) {
  const float* C_ri       = (const float*)d_in[0];
  const float* log_dt     = (const float*)d_in[1];
  const float* B_ri       = (const float*)d_in[2];
  const float* inv_A_real = (const float*)d_in[3];
  const float* A_imag     = (const float*)d_in[4];
  // d_in[5] is the L scalar on device; derive host-side values from sizes
  // instead (graph-capture safe: no sync copies).
  const int Hn    = in_sizes[1];              // H from log_dt
  const int n_ssm = in_sizes[3] / SSD_N2;     // from inv_A_real = (n_ssm, N2)
  const int L     = out_size / Hn;            // K is (H, L)

  float* Kout = (float*)d_out;
  (void)d_ws; (void)ws_size; (void)n_in;

  dim3 grid(Hn);
  dim3 block(128);  // 4 wave32 per head: each wave owns L/(256*4) output tiles
  s4d_vandermonde_wmma<<<grid, block, 0, stream>>>(
      C_ri, log_dt, B_ri, inv_A_real, A_imag, Kout, n_ssm, L);
}